// StyleTokenLayer_180388627041
// MI455X (gfx1250) — compile-verified
//
#include <hip/hip_runtime.h>
#include <hip/hip_bf16.h>

typedef __attribute__((ext_vector_type(2))) float v2f;
typedef __attribute__((ext_vector_type(8))) float v8f;

#define NUM_B   16
#define MAX_LEN 2048
#define C_EMO   30
#define HID     768
#define NT_N    (HID / 16)                    // 48 N-tiles
#define NT_M    ((NUM_B * MAX_LEN) / 16)      // 2048 M-tiles
#define WAVES_PER_BLOCK 8

__global__ __launch_bounds__(256)
void style_token_wmma_f32(const float* __restrict__ emo,
                          const int*   __restrict__ seqs_len,
                          const float* __restrict__ W,     // [C_EMO, HID]
                          float*       __restrict__ out,   // [NUM_B*MAX_LEN, HID]
                          int nrows)                       // TOTAL packed rows
{
    const int lane = threadIdx.x & 31;
    const int wave = threadIdx.x >> 5;
    const int tile = blockIdx.x * WAVES_PER_BLOCK + wave;

    const int tileN = tile % NT_N;
    const int tileM = tile / NT_N;

    const int rowBase = tileM * 16;            // flattened (b*MAX_LEN + l) base
    const int b       = rowBase / MAX_LEN;     // whole tile shares one batch
    const int posBase = rowBase % MAX_LEN;

    // Branchless per-batch packed offset: load all 16 lengths (4x int4),
    // fully unrolled masked prefix sum. No variable-trip loop, no branches.
    int4 L0 = ((const int4*)seqs_len)[0];
    int4 L1 = ((const int4*)seqs_len)[1];
    int4 L2 = ((const int4*)seqs_len)[2];
    int4 L3 = ((const int4*)seqs_len)[3];
    int len[NUM_B] = {L0.x, L0.y, L0.z, L0.w, L1.x, L1.y, L1.z, L1.w,
                      L2.x, L2.y, L2.z, L2.w, L3.x, L3.y, L3.z, L3.w};
    int offset = 0;
#pragma unroll
    for (int i = 0; i < NUM_B - 1; ++i) offset += (i < b) ? len[i] : 0;
    const int len_b = len[b];

    const int  m   = lane & 15;
    const bool hi  = lane >= 16;
    const int  pos = posBase + m;              // position within this sequence
    const bool vrow = pos < len_b;             // padding mask (reference: valid)

    // Clamp row index (reference: jnp.clip) so the load is always in-bounds,
    // then mask the *values* -> unconditional, clause-friendly loads.
    int ridx = offset + pos;
    ridx = ridx < (nrows - 1) ? ridx : (nrows - 1);
    const float* arow = emo + (long)ridx * C_EMO;

    // ---- A tile: 16 rows x 32 K (K=30 real, 2 zero-padded), 8 steps of K=4.
    // 32-bit A 16x4 layout: lanes 0-15 -> (k0,k0+1), lanes 16-31 -> (k0+2,k0+3).
    v2f a[8];
#pragma unroll
    for (int s = 0; s < 8; ++s) {
        const int kb = 4 * s + (hi ? 2 : 0);
        const int kc = kb < 28 ? kb : 28;      // in-bounds even pair (8B aligned)
        const bool kval = kb < C_EMO;          // only s==7 hi-lanes are false
        const float2 v = *(const float2*)(arow + kc);
        a[s].x = (vrow && kval) ? v.x : 0.f;   // v_cndmask, no exec branching
        a[s].y = (vrow && kval) ? v.y : 0.f;
    }

    // ---- B tile: 4x16 per step from W[k][tileN*16 + n], row-striped across lanes.
    const int col = tileN * 16 + m;
    v2f bm[8];
#pragma unroll
    for (int s = 0; s < 8; ++s) {
        const int kb = 4 * s + (hi ? 2 : 0);
        const int kc = kb < 28 ? kb : 28;
        const bool kval = kb < C_EMO;
        const float b0 = W[(size_t)kc       * HID + col];
        const float b1 = W[(size_t)(kc + 1) * HID + col];
        bm[s].x = kval ? b0 : 0.f;
        bm[s].y = kval ? b1 : 0.f;
    }

    // ---- 8 chained f32 WMMAs: C += A(16x4) x B(4x16)
    v8f c = {};
#pragma unroll
    for (int s = 0; s < 8; ++s) {
        c = __builtin_amdgcn_wmma_f32_16x16x4_f32(
                /*neg_a=*/false, a[s], /*neg_b=*/false, bm[s],
                /*c_mod=*/(short)0, c, /*reuse_a=*/false, /*reuse_b=*/false);
    }

    // ---- Store C/D: VGPR v, lane -> row = v + (hi?8:0), col = lane&15.
    // Streaming output written once, never re-read: non-temporal stores
    // keep the reused 92KB weight tile resident in cache.
#pragma unroll
    for (int v = 0; v < 8; ++v) {
        const int r = rowBase + v + (hi ? 8 : 0);
        __builtin_nontemporal_store(c[v], &out[(size_t)r * HID + col]);
    }
}

extern "C" void kernel_launch(void* const* d_in, const int* in_sizes, int n_in,
                              void* d_out, int out_size, void* d_ws, size_t ws_size,
                              hipStream_t stream) {
    const float* emo  = (const float*)d_in[0];   // [TOTAL, 30] f32
    const int*   sl   = (const int*)  d_in[1];   // [16] i32
    const float* W    = (const float*)d_in[2];   // [1, 30, 768] f32
    float*       out  = (float*)d_out;           // [16, 2048, 768] f32
    const int nrows   = in_sizes[0] / C_EMO;     // TOTAL = 16384

    const int totalTiles = NT_M * NT_N;                          // 98304
    const int blocks = totalTiles / WAVES_PER_BLOCK;             // 12288
    hipLaunchKernelGGL(style_token_wmma_f32, dim3(blocks), dim3(256), 0, stream,
                       emo, sl, W, out, nrows);
}